// Model_35648228556884
// MI455X (gfx1250) — compile-verified
//
#include <hip/hip_runtime.h>
#include <hip/hip_bf16.h>

// Problem constants from the reference
constexpr int B_ = 32;
constexpr int M_ = 24;
constexpr int R_ = 384;
constexpr int D_ = 2048;
constexpr int H_ = 16;
constexpr float NEGV = -100000000.0f;

constexpr int PROJ_WAVES = 8;   // waves per block in the WMMA projection kernel

typedef __attribute__((ext_vector_type(2))) float v2f;
typedef __attribute__((ext_vector_type(4))) float v4f;
typedef __attribute__((ext_vector_type(8))) float v8f;

// ---------------------------------------------------------------------------
// Kernel 1: h0 = region @ t0_w.T + t0_b ; h1 = region @ t1_w.T + t1_b
//
// Full-f32 WMMA (V_WMMA_F32_16X16X4_F32). One block = 8 waves cooperating on
// one 16-row output tile; each wave owns K/8 = 256 of the K=2048 reduction,
// then partial 16x16 accumulators are summed across waves through LDS.
//
// Per 8-K step each lane does one B128 load per matrix; lane l covers
// K = k + (l>>4)*4 .. +3. wmma#1 consumes vector elems {0,1}
// (K set {k,k+1} u {k+4,k+5}), wmma#2 consumes elems {2,3}
// (K set {k+2,k+3} u {k+6,k+7}) — every K counted once, and A/B use the
// same lane<->K-slot mapping so the contraction is consistent.
//
// D (16x16 f32) layout: VGPR v, lane l -> (m = v + (l>>4)*8, n = l&15).
// ---------------------------------------------------------------------------
__global__ __launch_bounds__(32 * PROJ_WAVES) void proj_wmma_kernel(
    const float* __restrict__ region,
    const float* __restrict__ t0_w, const float* __restrict__ t0_b,
    const float* __restrict__ t1_w, const float* __restrict__ t1_b,
    float* __restrict__ h0, float* __restrict__ h1)
{
    __shared__ float red0[PROJ_WAVES][32 * 8];
    __shared__ float red1[PROJ_WAVES][32 * 8];

    const int tid  = threadIdx.x;
    const int lane = tid & 31;
    const int wv   = tid >> 5;
    const int mn   = lane & 15;           // row (A) / col (B)
    const int kq   = (lane >> 4) * 4;     // K sub-offset within the 8-chunk
    const long tile = blockIdx.x;         // 0 .. B*R/16 - 1
    const long row  = tile * 16 + mn;     // global row in [0, B*R)

    const int kPer = D_ / PROJ_WAVES;     // 256 K per wave
    const int k0   = wv * kPer;

    const float* aRow  = region + row * (long)D_ + kq;
    const float* b0Row = t0_w + (long)mn * D_ + kq;   // B[k][n] = tN_w[n][k]
    const float* b1Row = t1_w + (long)mn * D_ + kq;

    v8f acc0 = {};
    v8f acc1 = {};

#pragma unroll 4
    for (int k = k0; k < k0 + kPer; k += 8) {
        const v4f a  = *(const v4f*)(aRow  + k);
        const v4f b0 = *(const v4f*)(b0Row + k);
        const v4f b1 = *(const v4f*)(b1Row + k);
        const v2f alo = {a[0], a[1]},  ahi = {a[2], a[3]};
        const v2f b0lo = {b0[0], b0[1]}, b0hi = {b0[2], b0[3]};
        const v2f b1lo = {b1[0], b1[1]}, b1hi = {b1[2], b1[3]};
        acc0 = __builtin_amdgcn_wmma_f32_16x16x4_f32(
            false, alo, false, b0lo, (short)0, acc0, false, false);
        acc0 = __builtin_amdgcn_wmma_f32_16x16x4_f32(
            false, ahi, false, b0hi, (short)0, acc0, false, false);
        acc1 = __builtin_amdgcn_wmma_f32_16x16x4_f32(
            false, alo, false, b1lo, (short)0, acc1, false, false);
        acc1 = __builtin_amdgcn_wmma_f32_16x16x4_f32(
            false, ahi, false, b1hi, (short)0, acc1, false, false);
    }

    // Cross-wave reduction of the partial 16x16 tiles through LDS.
#pragma unroll
    for (int v = 0; v < 8; ++v) {
        red0[wv][lane * 8 + v] = acc0[v];
        red1[wv][lane * 8 + v] = acc1[v];
    }
    __syncthreads();

    // 256 threads cover the 32x8 (lane, vgpr-slot) grid: one output each.
    {
        const int rlane = tid & 31;
        const int v     = tid >> 5;       // 0..7
        float s0 = 0.f, s1 = 0.f;
#pragma unroll
        for (int w = 0; w < PROJ_WAVES; ++w) {
            s0 += red0[w][rlane * 8 + v];
            s1 += red1[w][rlane * 8 + v];
        }
        const int n    = rlane & 15;
        const long orow = tile * 16 + v + ((rlane >> 4) << 3);
        h0[orow * H_ + n] = s0 + t0_b[n];
        h1[orow * H_ + n] = s1 + t1_b[n];
    }
}

// ---------------------------------------------------------------------------
// Kernel 2: tran[b,r,s] = sum_h relu(h0[b,r,h] + h1[b,s,h]) * tr_w[h] + tr_b
// 18.9 MB output -> L2-resident for the scan kernel.
// ---------------------------------------------------------------------------
__global__ __launch_bounds__(256) void tran_kernel(
    const float* __restrict__ h0, const float* __restrict__ h1,
    const float* __restrict__ tr_w, const float* __restrict__ tr_b,
    float* __restrict__ tran)
{
    const long idx = (long)blockIdx.x * blockDim.x + threadIdx.x;
    if (idx >= (long)B_ * R_ * R_) return;
    const int  s = (int)(idx % R_);
    const long t = idx / R_;
    const int  r = (int)(t % R_);
    const int  b = (int)(t / R_);

    const float4* p0 = (const float4*)(h0 + ((long)b * R_ + r) * H_);
    const float4* p1 = (const float4*)(h1 + ((long)b * R_ + s) * H_);
    const float4* w4 = (const float4*)tr_w;

    float acc = tr_b[0];
#pragma unroll
    for (int q = 0; q < H_ / 4; ++q) {
        const float4 x = p0[q];
        const float4 y = p1[q];
        const float4 w = w4[q];
        float v;
        v = x.x + y.x; acc += (v > 0.f ? v : 0.f) * w.x;
        v = x.y + y.y; acc += (v > 0.f ? v : 0.f) * w.y;
        v = x.z + y.z; acc += (v > 0.f ? v : 0.f) * w.z;
        v = x.w + y.w; acc += (v > 0.f ? v : 0.f) * w.w;
    }
    tran[idx] = acc;
}

// ---------------------------------------------------------------------------
// Kernel 3: the sequential scan over mentions. One workgroup per batch b,
// 384 threads (one per region r). alpha/score/rprev live in LDS; barriers
// between steps. Inactive steps skipped (uniform branch), matching the
// reference's masked-where semantics exactly.
// ---------------------------------------------------------------------------
__global__ __launch_bounds__(R_) void scan_kernel(
    const float* __restrict__ aff, const float* __restrict__ aff_gt,
    const int* __restrict__ n_mentions, const int* __restrict__ n_regions,
    const float* __restrict__ tran, float* __restrict__ partial)
{
    __shared__ float alpha[R_];
    __shared__ float score[R_];
    __shared__ float rprev[R_];
    __shared__ float ssum[M_];

    const int b = blockIdx.x;
    const int r = threadIdx.x;
    const int nreg = n_regions[b];
    const int nmen = n_mentions[b];

    if (r < M_) ssum[r] = 0.f;
    __syncthreads();

    // Per-mention reward normalizers: ssum[m] = sum_r masked aff_gt
    for (int m = 0; m < M_; ++m) {
        const float g = aff_gt[((long)(b * M_ + m)) * R_ + r];
        const float a = (g >= 0.5f && r < nreg && m < nmen) ? g : 0.f;
        if (a > 0.f) atomicAdd(&ssum[m], a);   // ds_add_f32
    }

    alpha[r] = (r < nreg) ? 0.f : NEGV;
    score[r] = 0.f;
    rprev[r] = 0.f;
    __syncthreads();

    bool hasp = false;                         // uniform across block
    for (int m = 0; m < M_; ++m) {
        const float sm = ssum[m];
        const bool act = (m < nmen) && (sm > 0.f);
        if (!act) continue;                    // nothing changes this step

        float alpha_t = 0.f, score_t = 0.f;
        if (hasp) {
            const float* trow = tran + ((long)b * R_ + r) * (long)R_;
            float mx = -3.0e38f;
            for (int s = 0; s < R_; ++s) {
                const float v = alpha[s] + trow[s];
                mx = fmaxf(mx, v);
            }
            float sum = 0.f;
            float sc  = 0.f;
            for (int s = 0; s < R_; ++s) {
                const float tv = trow[s];
                sum += __expf(alpha[s] + tv - mx);
                sc  += (score[s] + tv) * rprev[s];
            }
            alpha_t = mx + __logf(sum);
            score_t = sc;
        }

        const float e = aff[((long)(b * M_ + m)) * R_ + r];
        const float g = aff_gt[((long)(b * M_ + m)) * R_ + r];
        const float a = (g >= 0.5f && r < nreg) ? g : 0.f;  // m<nmen given act
        const float rw = a / sm;                            // sm > 0 given act
        const float rl = (rw > 0.f) ? __logf(rw) : NEGV;

        const float na = (hasp ? alpha_t : alpha[r]) + e;
        const float ns = (hasp ? score_t : score[r]) + e - rl;

        __syncthreads();        // old alpha/score/rprev fully consumed
        alpha[r] = na;
        score[r] = ns;
        rprev[r] = rw;
        __syncthreads();
        hasp = true;
    }

    __syncthreads();
    if (r == 0) {
        float out = 0.f;
        if (hasp) {
            float mx = -3.0e38f;
            for (int s = 0; s < R_; ++s) mx = fmaxf(mx, alpha[s]);
            float sum = 0.f, gold = 0.f;
            for (int s = 0; s < R_; ++s) {
                sum  += __expf(alpha[s] - mx);
                gold += score[s] * rprev[s];
            }
            out = (mx + __logf(sum)) - gold;
        }
        partial[b] = out;
    }
}

// ---------------------------------------------------------------------------
// Kernel 4: final reduction over batches + division by denom.
// ---------------------------------------------------------------------------
__global__ void finalize_kernel(const float* __restrict__ partial,
                                const int* __restrict__ n_mentions,
                                float* __restrict__ out)
{
    if (threadIdx.x == 0 && blockIdx.x == 0) {
        float t = 0.f;
        int d = 0;
        for (int b = 0; b < B_; ++b) {
            t += partial[b];
            d += n_mentions[b];
        }
        if (d < 1) d = 1;
        out[0] = t / (float)d;
    }
}

extern "C" void kernel_launch(void* const* d_in, const int* in_sizes, int n_in,
                              void* d_out, int out_size, void* d_ws, size_t ws_size,
                              hipStream_t stream)
{
    const float* region     = (const float*)d_in[0];   // (B,R,D)
    const float* aff        = (const float*)d_in[1];   // (B,M,R,1)
    const float* aff_gt     = (const float*)d_in[2];   // (B,M,R,1)
    const int*   n_mentions = (const int*)  d_in[3];   // (B,)
    const int*   n_regions  = (const int*)  d_in[4];   // (B,)
    const float* t0_w       = (const float*)d_in[5];   // (H,D)
    const float* t0_b       = (const float*)d_in[6];   // (H,)
    const float* t1_w       = (const float*)d_in[7];   // (H,D)
    const float* t1_b       = (const float*)d_in[8];   // (H,)
    const float* tr_w       = (const float*)d_in[9];   // (1,H)
    const float* tr_b       = (const float*)d_in[10];  // (1,)
    float* out = (float*)d_out;

    // Workspace layout
    char* base = (char*)d_ws;
    const size_t tranBytes = (size_t)B_ * R_ * R_ * sizeof(float);  // 18.9 MB
    const size_t hBytes    = (size_t)B_ * R_ * H_ * sizeof(float);  // 0.79 MB
    float* tran    = (float*)(base);
    float* h0      = (float*)(base + tranBytes);
    float* h1      = (float*)(base + tranBytes + hBytes);
    float* partial = (float*)(base + tranBytes + 2 * hBytes);

    // 1) Projections via f32 WMMA: 8 waves per tile splitting K, LDS reduce.
    proj_wmma_kernel<<<(B_ * R_) / 16, 32 * PROJ_WAVES, 0, stream>>>(
        region, t0_w, t0_b, t1_w, t1_b, h0, h1);

    // 2) Transition matrix (L2-resident scratch)
    {
        const long total = (long)B_ * R_ * R_;
        const int  blk = 256;
        const int  grid = (int)((total + blk - 1) / blk);
        tran_kernel<<<grid, blk, 0, stream>>>(h0, h1, tr_w, tr_b, tran);
    }

    // 3) Sequential mention scan: one workgroup per batch
    scan_kernel<<<B_, R_, 0, stream>>>(aff, aff_gt, n_mentions, n_regions,
                                       tran, partial);

    // 4) Final scalar
    finalize_kernel<<<1, 32, 0, stream>>>(partial, n_mentions, out);
}